// LandmarkLoss_75771813036236
// MI455X (gfx1250) — compile-verified
//
#include <hip/hip_runtime.h>
#include <hip/hip_bf16.h>

typedef __attribute__((ext_vector_type(2))) float v2f;
typedef __attribute__((ext_vector_type(8))) float v8f;

// ---------------------------------------------------------------------------
// Wave-wide sum of the 64 f32 values occupying the A-matrix slots of a
// V_WMMA_F32_16X16X4_F32 (16x4 A striped 2 floats/lane across the wave32).
// D = A x ones(4x16): D[i][j] = rowsum_i for every column j. Each lane holds
// 8 D values = sum over half the rows; adding the swapx16-shuffled value
// yields the full 64-element sum in every lane. EXEC must be all-1s in the
// calling wave (callers guarantee a full 32-lane wave).
// ---------------------------------------------------------------------------
__device__ __forceinline__ float wave_sum64_wmma(v2f a) {
  v2f b;
  b[0] = 1.0f;
  b[1] = 1.0f;
  v8f c = {};
  // 8 args: (neg_a, A, neg_b, B, c_mod, C, reuse_a, reuse_b)
  v8f d = __builtin_amdgcn_wmma_f32_16x16x4_f32(false, a, false, b,
                                                (short)0, c, false, false);
  float s = ((d[0] + d[1]) + (d[2] + d[3])) + ((d[4] + d[5]) + (d[6] + d[7]));
  s += __shfl_xor(s, 16, 32);  // combine row halves (lanes 0-15 vs 16-31)
  return s;
}

// ---------------------------------------------------------------------------
// One block per image. LDS bitmask (256x256 bits = 8KB) deduplicates
// overlapping 2x2 landmark patches via ds_or_rtn_b32; only the lane that
// first sets a bit gathers |gen-tar| over the 3 channels for that pixel.
// ---------------------------------------------------------------------------
#define HW 65536  // 256*256
#define NLM 68
#define NPIX 272  // 68 landmarks * 4 corner pixels

__global__ __launch_bounds__(256) void lm_loss_main(
    const float* __restrict__ gen, const float* __restrict__ tar,
    const float* __restrict__ lm, float* __restrict__ partial) {
  __shared__ unsigned int maskbits[2048];  // 65536 bits
  __shared__ float red[256];

  const int tid = threadIdx.x;
  const int img = blockIdx.x;

#pragma unroll
  for (int i = 0; i < 8; ++i) maskbits[tid + 256 * i] = 0u;
  __syncthreads();

  float s = 0.0f;
  if (tid < NPIX) {
    const int l = tid >> 2;   // landmark
    const int c = tid & 3;    // corner within 2x2 patch
    const float* lmb = lm + img * (2 + 2 * NLM);
    // astype(int32) truncates toward zero, then clip to [1,254]
    int x = (int)lmb[2 + l];
    int y = (int)lmb[2 + NLM + l];
    x = min(max(x, 1), 254);
    y = min(max(y, 1), 254);
    const int row = x - 1 + (c >> 1);
    const int col = y - 1 + (c & 1);
    const int pos = (row << 8) | col;
    const unsigned int bit = 1u << (pos & 31);
    const unsigned int old = atomicOr(&maskbits[pos >> 5], bit);  // ds_or_rtn
    if (!(old & bit)) {
      const size_t base = (size_t)img * 3 * HW + (size_t)pos;
#pragma unroll
      for (int ch = 0; ch < 3; ++ch) {
        const float g = gen[base + (size_t)ch * HW];
        const float t = tar[base + (size_t)ch * HW];
        s += fabsf(g - t);
      }
    }
  }

  red[tid] = s;
  __syncthreads();

  if (tid < 32) {  // wave 0 only: full EXEC for the WMMA reduce
    float v = 0.0f;
#pragma unroll
    for (int k = 0; k < 8; ++k) v += red[tid * 8 + k];  // fixed order
    v2f a;
    a[0] = v;
    a[1] = 0.0f;
    const float tot = wave_sum64_wmma(a);
    if (tid == 0) partial[img] = tot;
  }
}

// ---------------------------------------------------------------------------
// Single wave: 64 per-image partials fill the 16x4 A matrix exactly
// (2 values per lane); one WMMA reduce, lane 0 scales and writes the mean.
// ---------------------------------------------------------------------------
__global__ __launch_bounds__(32) void lm_loss_final(
    const float* __restrict__ partial, float* __restrict__ out) {
  const int lane = threadIdx.x;
  v2f a;
  a[0] = partial[2 * lane];
  a[1] = partial[2 * lane + 1];
  const float tot = wave_sum64_wmma(a);
  if (lane == 0) {
    out[0] = tot * (1.0f / 12582912.0f);  // / (64*3*256*256)
  }
}

extern "C" void kernel_launch(void* const* d_in, const int* in_sizes, int n_in,
                              void* d_out, int out_size, void* d_ws,
                              size_t ws_size, hipStream_t stream) {
  const float* gen = (const float*)d_in[0];  // (64,3,256,256) f32
  const float* tar = (const float*)d_in[1];  // (64,3,256,256) f32
  const float* lm = (const float*)d_in[2];   // (64,1,138)     f32
  float* out = (float*)d_out;                // scalar f32
  float* partial = (float*)d_ws;             // 64 f32 partials

  lm_loss_main<<<64, 256, 0, stream>>>(gen, tar, lm, partial);
  lm_loss_final<<<1, 32, 0, stream>>>(partial, out);
}